// CPNContextMatchingComplete_26731876450691
// MI455X (gfx1250) — compile-verified
//
#include <hip/hip_runtime.h>
#include <math.h>

typedef float v2f __attribute__((ext_vector_type(2)));
typedef float v8f __attribute__((ext_vector_type(8)));

// Problem constants
constexpr int B = 8, CC = 128, F = 5, R = 4;
constexpr int H = 64, W = 64, HW = H * W;          // 4096
constexpr int HW4 = HW / 4;                        // 1024 float4 groups
constexpr int HI = 256, WI = 256, HWI = HI * WI;   // 65536

// Workspace layout (in floats)
constexpr int WS_VTR  = 0;                 // B*HW          binarized vt_r
constexpr int WS_VR   = WS_VTR + B * HW;   // B*R*HW        binarized vr
constexpr int WS_VSUM = WS_VR + B * R * HW;// B*R
constexpr int WS_GS   = WS_VSUM + B * R;   // B*R           atomic accumulators
constexpr int WS_CM   = WS_GS + B * R;     // B*R*HW        c_match (channel-independent)

// Output layout (in floats): concat(out1, out2, out3)
constexpr int OUT1 = 0;                          // (B, CC+1, H, W)
constexpr int OUT2 = B * (CC + 1) * HW;          // (B, 1, H, W)
constexpr int OUT3 = OUT2 + B * HW;              // (B, 3, F, HI, WI)

// ---------------------------------------------------------------- init
__global__ void k_init_gs(float* ws) {
  int i = threadIdx.x;
  if (i < B * R) ws[WS_GS + i] = 0.0f;
}

// ------------------------------------------------- antialiased downsample + binarize
// jax.image.resize(bilinear, antialias) 256->64: triangle filter, 8 taps/axis,
// out-of-range taps dropped with renormalization (matches jax weight matrix).
__global__ void k_masks(const float* __restrict__ vt, const float* __restrict__ va,
                        float* __restrict__ ws) {
  int idx = blockIdx.x * blockDim.x + threadIdx.x;
  const int total = B * HW + B * R * HW;
  if (idx >= total) return;
  const float wtab[8] = {0.125f, 0.375f, 0.625f, 0.875f,
                         0.875f, 0.625f, 0.375f, 0.125f};
  const float* src;
  float* dst;
  int oy, ox;
  if (idx < B * HW) {
    int b = idx / HW, p = idx % HW;
    oy = p / W; ox = p % W;
    src = vt + b * HWI;
    dst = ws + WS_VTR + idx;
  } else {
    int j = idx - B * HW;
    int b = j / (R * HW), rr = (j / HW) % R, p = j % HW;
    oy = p / W; ox = p % W;
    src = va + (b * R + rr) * HWI;
    dst = ws + WS_VR + j;
  }
  float acc = 0.0f, wsum = 0.0f;
  for (int ty = 0; ty < 8; ++ty) {
    int iy = 4 * oy - 2 + ty;
    if (iy < 0 || iy >= HI) continue;
    float wy = wtab[ty];
    const float* row = src + iy * WI;
    for (int tx = 0; tx < 8; ++tx) {
      int ix = 4 * ox - 2 + tx;
      if (ix < 0 || ix >= WI) continue;
      float wv = wy * wtab[tx];
      acc += wv * row[ix];
      wsum += wv;
    }
  }
  *dst = (acc / wsum > 0.5f) ? 1.0f : 0.0f;
}

// ---------------------------------------------------------------- v_sum
__global__ void k_vsum(float* ws) {
  __shared__ float red[256];
  int br = blockIdx.x;             // b*R + r
  int b = br / R;
  const float* vtr = ws + WS_VTR + b * HW;
  const float* vr  = ws + WS_VR + br * HW;
  float s = 0.0f;
  for (int p = threadIdx.x; p < HW; p += 256) s += vtr[p] * vr[p];
  red[threadIdx.x] = s;
  __syncthreads();
  for (int k = 128; k > 0; k >>= 1) {
    if (threadIdx.x < k) red[threadIdx.x] += red[threadIdx.x + k];
    __syncthreads();
  }
  if (threadIdx.x == 0) ws[WS_VSUM + br] = red[0];
}

// ------------------------------------------- gs reduction with WMMA wave-reduce
// gs_acc[b,r] += sum_{c,hw} vt_r*vr * f0 * f_{r+1}
__global__ void __launch_bounds__(256)
k_gs(const float* __restrict__ cf, const float* __restrict__ ws,
     float* __restrict__ gs_acc) {
  int b = blockIdx.y;
  const float4* f4  = (const float4*)cf;
  const float4* vt4 = (const float4*)(ws + WS_VTR);
  const float4* vr4 = (const float4*)(ws + WS_VR);
  float loc[R] = {0.f, 0.f, 0.f, 0.f};
  const int NG = CC * HW4;  // 131072 float4 groups per batch
  for (int g = blockIdx.x * 256 + threadIdx.x; g < NG; g += gridDim.x * 256) {
    int c = g >> 10, hw4 = g & (HW4 - 1);
    int base = ((b * CC + c) * F) * HW4 + hw4;
    __builtin_prefetch(&f4[base + gridDim.x * 256], 0, 1);
    float4 f0 = f4[base];
    float4 vt = vt4[b * HW4 + hw4];
    float4 m0; m0.x = vt.x * f0.x; m0.y = vt.y * f0.y;
    m0.z = vt.z * f0.z; m0.w = vt.w * f0.w;
#pragma unroll
    for (int r = 0; r < R; ++r) {
      float4 fr = f4[base + (1 + r) * HW4];
      float4 vr = vr4[(b * R + r) * HW4 + hw4];
      loc[r] += m0.x * vr.x * fr.x + m0.y * vr.y * fr.y +
                m0.z * vr.z * fr.z + m0.w * vr.w * fr.w;
    }
  }
  // Wave-level reduction: V_WMMA_F32_16X16X4_F32 with B = ones row-sums the
  // A matrix (lane sums live in K0 for lanes 0-15 and K2 for lanes 16-31).
  // D[m,n] = laneSum_m + laneSum_{m+16}; lane n holds M=0..7 (n<16) / 8..15.
  int lane = threadIdx.x & 31;
#pragma unroll
  for (int r = 0; r < R; ++r) {
    v2f a; a[0] = loc[r]; a[1] = 0.0f;
    v2f bo; bo[0] = 1.0f; bo[1] = 1.0f;
    v8f cacc = {0.f, 0.f, 0.f, 0.f, 0.f, 0.f, 0.f, 0.f};
    cacc = __builtin_amdgcn_wmma_f32_16x16x4_f32(
        false, a, false, bo, (short)0, cacc, false, false);
    float v = cacc[0] + cacc[1] + cacc[2] + cacc[3] +
              cacc[4] + cacc[5] + cacc[6] + cacc[7];
    v += __shfl_xor(v, 16);   // combine the two half-wave partial sums
    if (lane == 0) atomicAdd(&gs_acc[b * R + r], v);
  }
}

// ------------------------------------- masked softmax -> c_match, c_mask
__global__ void k_cmatch(float* __restrict__ ws, float* __restrict__ out) {
  int idx = blockIdx.x * blockDim.x + threadIdx.x;   // over B*HW
  if (idx >= B * HW) return;
  int b = idx / HW, p = idx % HW;
  float vt = ws[WS_VTR + idx];
  float g[R], m[R];
#pragma unroll
  for (int r = 0; r < R; ++r) {
    float vs = ws[WS_VSUM + b * R + r];
    bool vz = vs < 1e-4f;
    float denom = (vs + (vz ? 1.0f : 0.0f)) * (float)CC;
    g[r] = vz ? 0.0f : ws[WS_GS + b * R + r] / denom;
    m[r] = ws[WS_VR + (b * R + r) * HW + p];
  }
  float mx = fmaxf(fmaxf(g[0] * m[0], g[1] * m[1]),
                   fmaxf(g[2] * m[2], g[3] * m[3]));
  float e[R], s = 0.0f;
#pragma unroll
  for (int r = 0; r < R; ++r) {
    e[r] = expf(g[r] * m[r] - mx) * m[r];
    s += e[r];
  }
  s += (s < 1e-4f) ? 1.0f : 0.0f;
  float inv = 1.0f / s, keep = 1.0f - vt;
  float cmsum = 0.0f;
#pragma unroll
  for (int r = 0; r < R; ++r) {
    float cm = e[r] * inv * keep;
    ws[WS_CM + (b * R + r) * HW + p] = cm;
    cmsum += cm * m[r];
  }
  float cmask = 1.0f - cmsum;
  out[OUT1 + (b * (CC + 1) + CC) * HW + p] = cmask;   // channel 128 of output 1
  out[OUT2 + b * HW + p] = cmask;                     // output 2
}

// --------------------------------------------- decoder_input (big stream)
__global__ void __launch_bounds__(256)
k_decoder(const float* __restrict__ cf, const float* __restrict__ ws,
          float* __restrict__ out) {
  int idx = blockIdx.x * 256 + threadIdx.x;   // over B*CC*HW4
  if (idx >= B * CC * HW4) return;
  int hw4 = idx & (HW4 - 1), bc = idx >> 10;
  int b = bc / CC;
  const float4* f4  = (const float4*)cf;
  const float4* vt4 = (const float4*)(ws + WS_VTR);
  const float4* cm4 = (const float4*)(ws + WS_CM);
  int base = bc * F * HW4 + hw4;
  float4 f0 = f4[base];
  float4 vt = vt4[b * HW4 + hw4];
  float4 o;
  o.x = f0.x * vt.x; o.y = f0.y * vt.y; o.z = f0.z * vt.z; o.w = f0.w * vt.w;
#pragma unroll
  for (int r = 0; r < R; ++r) {
    float4 fr = f4[base + (1 + r) * HW4];
    float4 cm = cm4[(b * R + r) * HW4 + hw4];
    o.x += fr.x * cm.x; o.y += fr.y * cm.y;
    o.z += fr.z * cm.z; o.w += fr.w * cm.w;
  }
  ((float4*)out)[(b * (CC + 1) + (bc % CC)) * HW4 + hw4] = o;
}

// ---------------------------------------------------- ref_importance
__global__ void k_refimp(const float* __restrict__ vt, const float* __restrict__ ws,
                         float* __restrict__ out) {
  int idx = blockIdx.x * blockDim.x + threadIdx.x;   // over B*F*HWI
  if (idx >= B * F * HWI) return;
  int p = idx % HWI, bf = idx / HWI;
  int ft = bf % F, b = bf / F;
  int hi = p / WI, wi = p % WI;
  const float* vtb = vt + b * HWI;
  // border = 3x3 max-dilation of (1 - v_t) strictly exceeds center
  float mc = 1.0f - vtb[p];
  float dil = mc;
  for (int dy = -1; dy <= 1; ++dy) {
    int y = hi + dy;
    if (y < 0 || y >= HI) continue;
    for (int dx = -1; dx <= 1; ++dx) {
      int x = wi + dx;
      if (x < 0 || x >= WI) continue;
      dil = fmaxf(dil, 1.0f - vtb[y * WI + x]);
    }
  }
  bool border = (dil - mc) > 0.0f;
  float base;
  if (ft == 2) {
    base = vtb[p];
  } else {
    int r = (ft < 2) ? ft : ft - 1;
    const float* cm = ws + WS_CM + (b * R + r) * HW;
    float xf = wi * 0.25f - 0.375f;
    float yf = hi * 0.25f - 0.375f;
    int x0 = (int)floorf(xf), y0 = (int)floorf(yf);
    float tx = xf - (float)x0, ty = yf - (float)y0;
    int x0c = min(max(x0, 0), W - 1), x1c = min(max(x0 + 1, 0), W - 1);
    int y0c = min(max(y0, 0), H - 1), y1c = min(max(y0 + 1, 0), H - 1);
    base = (1.0f - ty) * ((1.0f - tx) * cm[y0c * W + x0c] + tx * cm[y0c * W + x1c]) +
           ty * ((1.0f - tx) * cm[y1c * W + x0c] + tx * cm[y1c * W + x1c]);
  }
  float v0 = border ? 1.0f : base;
  float v12 = border ? 0.0f : base;
  int ob = OUT3 + ((b * 3 + 0) * F + ft) * HWI + p;
  out[ob] = v0;                 // red channel
  out[ob + F * HWI] = v12;      // green
  out[ob + 2 * F * HWI] = v12;  // blue
}

extern "C" void kernel_launch(void* const* d_in, const int* in_sizes, int n_in,
                              void* d_out, int out_size, void* d_ws, size_t ws_size,
                              hipStream_t stream) {
  const float* cf = (const float*)d_in[0];   // c_feats (8,128,5,64,64)
  const float* vt = (const float*)d_in[1];   // v_t     (8,1,256,256)
  const float* va = (const float*)d_in[2];   // v_aligned (8,1,4,256,256)
  float* out = (float*)d_out;
  float* ws = (float*)d_ws;

  k_init_gs<<<1, 64, 0, stream>>>(ws);

  int nmask = B * HW + B * R * HW;
  k_masks<<<(nmask + 255) / 256, 256, 0, stream>>>(vt, va, ws);

  k_vsum<<<B * R, 256, 0, stream>>>(ws);

  dim3 ggrid(128, B);
  k_gs<<<ggrid, 256, 0, stream>>>(cf, ws, ws + WS_GS);

  k_cmatch<<<(B * HW + 255) / 256, 256, 0, stream>>>(ws, out);

  k_decoder<<<(B * CC * HW4 + 255) / 256, 256, 0, stream>>>(cf, ws, out);

  k_refimp<<<(B * F * HWI + 255) / 256, 256, 0, stream>>>(vt, ws, out);
}